// SpikeFP32Linear_MultiPrecision_43860206027331
// MI455X (gfx1250) — compile-verified
//
#include <hip/hip_runtime.h>

// Problem dims (fixed by the reference): x[256,1024,32], w[1024,1024,32],
// out[256,1024,32]; GEMM is C[256,1024] = X[256,1024] * W[1024,1024]^T in FP32.
#define BATCH 256
#define INF   1024
#define OUTF  1024

typedef __attribute__((ext_vector_type(2))) float v2f;
typedef __attribute__((ext_vector_type(8))) float v8f;

// ---------------------------------------------------------------------------
// Stage 1: decode pulse bits -> IEEE FP32. Pulse index 0 = MSB (sign bit).
// Each thread rebuilds one float from 32 pulse floats (8 vectorized float4
// loads), writes the packed value to workspace. This stage carries almost all
// of the HBM traffic (160 MB of pulses) and is pure streaming.
// ---------------------------------------------------------------------------
__global__ void decode_pulses_kernel(const float* __restrict__ pulses,
                                     float* __restrict__ vals, int n) {
  int idx = blockIdx.x * blockDim.x + threadIdx.x;
  if (idx >= n) return;
  const float4* p = (const float4*)(pulses + (size_t)idx * 32);
  unsigned u = 0u;
#pragma unroll
  for (int q = 0; q < 8; ++q) {
    float4 v = p[q];
    int base = 31 - 4 * q;              // shift for pulse index 4q (MSB-first)
    u |= (v.x > 0.5f ? 1u : 0u) << base;
    u |= (v.y > 0.5f ? 1u : 0u) << (base - 1);
    u |= (v.z > 0.5f ? 1u : 0u) << (base - 2);
    u |= (v.w > 0.5f ? 1u : 0u) << (base - 3);
  }
  vals[idx] = __uint_as_float(u);
}

// ---------------------------------------------------------------------------
// Stage 2: FP32 GEMM on the matrix pipe: V_WMMA_F32_16X16X4_F32.
// One wave per 16x16 output tile; K walked 0..1023 in ascending steps of 4
// (matches the reference's left-to-right accumulation order at instruction
// granularity). A/B operand layout per CDNA5 ISA 7.12.2:
//   lane<16: holds K = k0+0,k0+1 ; lane>=16: K = k0+2,k0+3 (contiguous b64).
// Epilogue fused: re-encode each accumulator value to 32 pulse floats.
// Decoded matrices (5 MB) are L2-resident, so plain global loads suffice.
// ---------------------------------------------------------------------------
__global__ void __launch_bounds__(256, 2)
pulse_gemm_wmma_kernel(const float* __restrict__ xf,
                       const float* __restrict__ wf,
                       float* __restrict__ out_pulses) {
  const int lane = threadIdx.x & 31;
  const int wave = (blockIdx.x << 3) | (threadIdx.x >> 5);  // 8 waves / block
  const int tm   = wave >> 6;   // 0..15  (256/16 row tiles)
  const int tn   = wave & 63;   // 0..63  (1024/16 col tiles)
  const int half = lane >> 4;   // 0 or 1
  const int l15  = lane & 15;
  const int koff = half * 2;    // K sub-offset owned by this half-wave

  const float* arow = xf + (size_t)(tm * 16 + l15) * INF + koff;
  const float* brow = wf + (size_t)(tn * 16 + l15) * INF + koff;

  v8f acc = {};
  for (int k0 = 0; k0 < INF; k0 += 4) {
    v2f a = *(const v2f*)(arow + k0);
    v2f b = *(const v2f*)(brow + k0);
    acc = __builtin_amdgcn_wmma_f32_16x16x4_f32(
        /*neg_a=*/false, a, /*neg_b=*/false, b,
        /*c_mod=*/(short)0, acc, /*reuse_a=*/false, /*reuse_b=*/false);
  }

  // Fused encode: lane holds C[M = tm*16 + r + 8*half][N = tn*16 + l15] in
  // acc[r]. Emit 32 pulse floats per value as 8 aligned float4 stores
  // (base offset is a multiple of 128 B).
#pragma unroll
  for (int r = 0; r < 8; ++r) {
    const int m = tm * 16 + r + half * 8;   // batch index
    const int n = tn * 16 + l15;            // out-feature index
    const unsigned u = __float_as_uint(acc[r]);
    float* dst = out_pulses + ((size_t)m * OUTF + n) * 32;
#pragma unroll
    for (int i = 0; i < 32; i += 4) {
      float4 v;
      v.x = (float)((u >> (31 - i)) & 1u);
      v.y = (float)((u >> (30 - i)) & 1u);
      v.z = (float)((u >> (29 - i)) & 1u);
      v.w = (float)((u >> (28 - i)) & 1u);
      *(float4*)(dst + i) = v;
    }
  }
}

extern "C" void kernel_launch(void* const* d_in, const int* in_sizes, int n_in,
                              void* d_out, int out_size, void* d_ws, size_t ws_size,
                              hipStream_t stream) {
  const float* x_pulses = (const float*)d_in[0];   // [256,1024,32]
  const float* w_pulses = (const float*)d_in[1];   // [1024,1024,32]
  float* out = (float*)d_out;                      // [256,1024,32]

  // Workspace: decoded xf (1 MB) followed by decoded wf (4 MB).
  float* xf = (float*)d_ws;
  float* wf = xf + (size_t)BATCH * INF;

  const int nx = BATCH * INF;   // 262144
  const int nw = OUTF * INF;    // 1048576
  decode_pulses_kernel<<<nx / 256, 256, 0, stream>>>(x_pulses, xf, nx);
  decode_pulses_kernel<<<nw / 256, 256, 0, stream>>>(w_pulses, wf, nw);

  // 1024 output tiles -> 1024 waves -> 128 blocks of 256 threads (wave32).
  pulse_gemm_wmma_kernel<<<128, 256, 0, stream>>>(xf, wf, out);
}